// EtaWeights_28767690948964
// MI455X (gfx1250) — compile-verified
//
#include <hip/hip_runtime.h>

// Elementwise eta-weights: out[i] = loss[i] > eta ? loss[i]*mask*eta
//                                                 : -loss[i]/eta + 1
// HBM-bandwidth-bound stream (256 MB traffic @ 23.3 TB/s => ~11.5 us floor).
// b128 non-temporal loads/stores, 2 groups (32B) per lane to amortize the
// per-wave prologue, single v_rcp_f32 for the uniform reciprocal.

typedef float v4f __attribute__((ext_vector_type(4)));

__global__ __launch_bounds__(256) void eta_weights_kernel(
    const float* __restrict__ loss,
    const float* __restrict__ eta_p,
    const float* __restrict__ mask_p,
    float* __restrict__ out,
    int n,    // total elements
    int n8)   // number of full groups of 8 floats (2 x float4)
{
    // Uniform parameters (s_load + one-time math).
    const float eta = eta_p[0];
    const float me  = mask_p[0] * eta;              // hot-branch multiplier
    const float nie = -__builtin_amdgcn_rcpf(eta);  // single v_rcp_f32

    const int i = blockIdx.x * blockDim.x + threadIdx.x;

    if (i < n8) {
        const v4f* lp = (const v4f*)loss;
        v4f*       op = (v4f*)out;
        const int  g  = i * 2;                      // first float4 group

        // 2x global_load_b128 TH=NT (pure stream, no L2 reuse)
        v4f l0 = __builtin_nontemporal_load(lp + g);
        v4f l1 = __builtin_nontemporal_load(lp + g + 1);

        v4f r0, r1;
#pragma unroll
        for (int j = 0; j < 4; ++j) {
            float x0 = l0[j], x1 = l1[j];
            r0[j] = (x0 > eta) ? (x0 * me) : __builtin_fmaf(x0, nie, 1.0f);
            r1[j] = (x1 > eta) ? (x1 * me) : __builtin_fmaf(x1, nie, 1.0f);
        }

        // 2x global_store_b128 TH=NT
        __builtin_nontemporal_store(r0, op + g);
        __builtin_nontemporal_store(r1, op + g + 1);
    }

    // Scalar tail (at most 7 elements); lane 0 of block 0 handles it.
    if (i == 0) {
        for (int k = n8 * 8; k < n; ++k) {
            float x = loss[k];
            out[k] = (x > eta) ? (x * me) : __builtin_fmaf(x, nie, 1.0f);
        }
    }
}

extern "C" void kernel_launch(void* const* d_in, const int* in_sizes, int n_in,
                              void* d_out, int out_size, void* d_ws, size_t ws_size,
                              hipStream_t stream) {
    (void)n_in; (void)d_ws; (void)ws_size; (void)out_size;

    const float* loss = (const float*)d_in[0];
    const float* eta  = (const float*)d_in[1];
    const float* mask = (const float*)d_in[2];
    float*       out  = (float*)d_out;

    const int n  = in_sizes[0];
    const int n8 = n >> 3;                 // groups of 8 floats per thread

    const int threads = 256;               // 8 wave32 per block
    const int blocks  = (n8 + threads - 1) / threads;

    eta_weights_kernel<<<blocks, threads, 0, stream>>>(loss, eta, mask, out, n, n8);
}